// DeLiteGNNStage_43121471652568
// MI455X (gfx1250) — compile-verified
//
#include <hip/hip_runtime.h>
#include <hip/hip_bf16.h>

typedef __attribute__((ext_vector_type(2))) float v2f;
typedef __attribute__((ext_vector_type(8))) float v8f;

#define DFEAT 128

__device__ __forceinline__ void atomAddF(float* p, float v) {
  // relaxed, device(agent) scope -> global_atomic_add_f32
  __hip_atomic_fetch_add(p, v, __ATOMIC_RELAXED, __HIP_MEMORY_SCOPE_AGENT);
}

// ---------------------------------------------------------------------------
// Per-hop degree counts: deg[k*n + v] = #edges of hop k with src==v (first 4*n)
//                        deg[(4+k)*n + v] = same for dst                (next 4*n)
// ---------------------------------------------------------------------------
__global__ __launch_bounds__(256) void degree_kernel(
    const int* __restrict__ src, const int* __restrict__ dst,
    const int* __restrict__ attr, float* __restrict__ deg, int n, int nedges) {
  int e = blockIdx.x * blockDim.x + threadIdx.x;
  if (e >= nedges) return;
  int k = attr[e] - 1;                 // 0..3
  atomAddF(&deg[(size_t)k * n + src[e]], 1.0f);
  atomAddF(&deg[(size_t)(4 + k) * n + dst[e]], 1.0f);
}

// ---------------------------------------------------------------------------
// coef[e] = invsqrt(deg_s) * invsqrt(deg_d) restricted to the edge's own hop
// ---------------------------------------------------------------------------
__global__ __launch_bounds__(256) void coef_kernel(
    const int* __restrict__ src, const int* __restrict__ dst,
    const int* __restrict__ attr, const float* __restrict__ deg,
    float* __restrict__ coef, int n, int nedges) {
  int e = blockIdx.x * blockDim.x + threadIdx.x;
  if (e >= nedges) return;
  int k = attr[e] - 1;
  float ds = deg[(size_t)k * n + src[e]];
  float dd = deg[(size_t)(4 + k) * n + dst[e]];
  float a = ds > 0.0f ? rsqrtf(fmaxf(ds, 1.0f)) : 0.0f;
  float b = dd > 0.0f ? rsqrtf(fmaxf(dd, 1.0f)) : 0.0f;
  coef[e] = a * b;
}

// ---------------------------------------------------------------------------
// H = scale * (X @ W),  X: n x 128, W: 128 x 128 row-major, scale = *nu_elem.
// One wave per 16x16 output tile. fp32 WMMA 16x16x4 (ISA 7.12.2 layouts):
//   A (16x4):  lanes 0-15 M=lane, v0=K0 v1=K1; lanes 16-31 M=lane-16, v0=K2 v1=K3
//   B (4x16):  v0 = row K0 (lanes 0-15) / K2 (lanes 16-31); v1 = K1 / K3
//   C/D (16x16): vgpr i: M = i + (lane>>4)*8, N = lane&15
// ---------------------------------------------------------------------------
__global__ __launch_bounds__(256) void gemm_nuw_wmma(
    const float* __restrict__ X, const float* __restrict__ Wm,
    const float* __restrict__ nu_elem, float* __restrict__ H, int n) {
  const int lane = threadIdx.x & 31;
  const int wave = blockIdx.x * (blockDim.x >> 5) + (threadIdx.x >> 5);
  const int tm = wave >> 3;          // 8 N-tiles of 16 cover D=128
  const int tn = wave & 7;
  if (tm * 16 >= n) return;

  const float scale = *nu_elem;
  const int row   = tm * 16 + (lane & 15);
  const int colB  = tn * 16 + (lane & 15);
  const int khalf = (lane >> 4) * 2;           // 0 or 2

  const float* xrow = X + (size_t)row * DFEAT + khalf;
  v8f c = {};
#pragma unroll 8
  for (int kk = 0; kk < DFEAT; kk += 4) {
    v2f a;
    a.x = xrow[kk];
    a.y = xrow[kk + 1];
    v2f b;
    b.x = Wm[(size_t)(kk + khalf) * DFEAT + colB];
    b.y = Wm[(size_t)(kk + khalf + 1) * DFEAT + colB];
    c = __builtin_amdgcn_wmma_f32_16x16x4_f32(
        /*neg_a=*/false, a, /*neg_b=*/false, b,
        /*c_mod=*/(short)0, c, /*reuse_a=*/false, /*reuse_b=*/false);
  }

  const int mBase = tm * 16 + (lane >> 4) * 8;
  const int nCol  = tn * 16 + (lane & 15);
#pragma unroll
  for (int i = 0; i < 8; ++i)
    H[(size_t)(mBase + i) * DFEAT + nCol] = c[i] * scale;
}

// ---------------------------------------------------------------------------
// Edge pass for hop k: acc[dst] += coef[e] * H[src]   (one wave per edge,
// each lane moves one float4 of the 128-float row; FP32 hw atomics)
// ---------------------------------------------------------------------------
__global__ __launch_bounds__(256) void edge_scatter(
    const float* __restrict__ H, const int* __restrict__ src,
    const int* __restrict__ dst, const int* __restrict__ attr,
    const float* __restrict__ coef, float* __restrict__ acc,
    int ksel, int nedges) {
  const int lane = threadIdx.x & 31;
  const long long wid =
      (long long)blockIdx.x * (blockDim.x >> 5) + (threadIdx.x >> 5);
  if (wid >= nedges) return;
  const int e = (int)wid;
  if (attr[e] != ksel) return;       // wave-uniform branch

  const float cw = coef[e];
  const int s = src[e];
  const int d = dst[e];
  const float4 v = ((const float4*)(H + (size_t)s * DFEAT))[lane];
  float* ap = acc + (size_t)d * DFEAT + lane * 4;
  atomAddF(ap + 0, v.x * cw);
  atomAddF(ap + 1, v.y * cw);
  atomAddF(ap + 2, v.z * cw);
  atomAddF(ap + 3, v.w * cw);
}

// ---------------------------------------------------------------------------
// xt = (prev + relu(acc)) / max(||.||2, 1e-12)   (one wave per node)
// ---------------------------------------------------------------------------
__global__ __launch_bounds__(256) void finalize_kernel(
    const float* __restrict__ prev, const float* __restrict__ acc,
    float* __restrict__ out, int n) {
  const int lane = threadIdx.x & 31;
  const int node = blockIdx.x * (blockDim.x >> 5) + (threadIdx.x >> 5);
  if (node >= n) return;

  const float4 p = ((const float4*)(prev + (size_t)node * DFEAT))[lane];
  const float4 a = ((const float4*)(acc + (size_t)node * DFEAT))[lane];
  float4 v;
  v.x = p.x + fmaxf(a.x, 0.0f);
  v.y = p.y + fmaxf(a.y, 0.0f);
  v.z = p.z + fmaxf(a.z, 0.0f);
  v.w = p.w + fmaxf(a.w, 0.0f);

  float ss = v.x * v.x + v.y * v.y + v.z * v.z + v.w * v.w;
#pragma unroll
  for (int m = 16; m >= 1; m >>= 1) ss += __shfl_xor(ss, m, 32);

  const float inv = 1.0f / fmaxf(sqrtf(ss), 1e-12f);
  float4 o;
  o.x = v.x * inv; o.y = v.y * inv; o.z = v.z * inv; o.w = v.w * inv;
  ((float4*)(out + (size_t)node * DFEAT))[lane] = o;
}

// ---------------------------------------------------------------------------
extern "C" void kernel_launch(void* const* d_in, const int* in_sizes, int n_in,
                              void* d_out, int out_size, void* d_ws, size_t ws_size,
                              hipStream_t stream) {
  const float* x    = (const float*)d_in[0];   // [N, 128]
  const int*   eix  = (const int*)d_in[1];     // [2, E]
  const int*   attr = (const int*)d_in[2];     // [E]
  const float* Wt   = (const float*)d_in[3];   // [T, 128, 128]
  const float* nu   = (const float*)d_in[4];   // [T, T]

  const int n = in_sizes[0] / DFEAT;           // 100000
  const int E = in_sizes[2];                   // 1.6M
  int T = 1;
  while (T * T < in_sizes[4]) ++T;             // 4

  const int* src = eix;
  const int* dst = eix + E;

  // workspace layout (floats): deg[8n] | coef[E] | H[n*D] | acc[n*D] | hist1..3
  float* ws    = (float*)d_ws;
  float* deg   = ws;
  float* coef  = deg + (size_t)8 * n;
  float* Hbuf  = coef + (size_t)E;
  float* acc   = Hbuf + (size_t)n * DFEAT;
  float* hist1 = acc + (size_t)n * DFEAT;
  float* hist2 = hist1 + (size_t)n * DFEAT;
  float* hist3 = hist2 + (size_t)n * DFEAT;
  const float* hist[5] = {x, hist1, hist2, hist3, (const float*)d_out};

  hipMemsetAsync(deg, 0, (size_t)8 * n * sizeof(float), stream);

  const int eb = (E + 255) / 256;
  degree_kernel<<<eb, 256, 0, stream>>>(src, dst, attr, deg, n, E);
  coef_kernel<<<eb, 256, 0, stream>>>(src, dst, attr, deg, coef, n, E);

  const int gemmBlocks = (n + 15) / 16;        // 8 waves/block = 8 N-tiles
  const int scatBlocks = (E + 7) / 8;          // 8 edges/block (wave each)
  const int finBlocks  = (n + 7) / 8;          // 8 nodes/block

  for (int t = 0; t < T; ++t) {
    hipMemsetAsync(acc, 0, (size_t)n * DFEAT * sizeof(float), stream);
    for (int k = 1; k <= t + 1; ++k) {
      const float* X   = hist[t + 1 - k];
      const float* Wk  = Wt + (size_t)(k - 1) * DFEAT * DFEAT;
      const float* nup = nu + (size_t)(k - 1) * T + t;
      gemm_nuw_wmma<<<gemmBlocks, 256, 0, stream>>>(X, Wk, nup, Hbuf, n);
      edge_scatter<<<scatBlocks, 256, 0, stream>>>(Hbuf, src, dst, attr, coef,
                                                   acc, k, E);
    }
    float* outT = (float*)hist[t + 1];
    finalize_kernel<<<finBlocks, 256, 0, stream>>>(hist[t], acc, outT, n);
  }
}